// DynamicEmbedding_42073499631937
// MI455X (gfx1250) — compile-verified
//
#include <hip/hip_runtime.h>
#include <math.h>

typedef __attribute__((ext_vector_type(2))) float v2f;
typedef __attribute__((ext_vector_type(8))) float v8f;

#define B_ 64
#define M_ 4096
#define D_ 256
#define NEG_MASK_VAL 1e-34f
#define LOG_MASK_EPS 1e-45f

// ---------------------------------------------------------------------------
// Kernel 1: hw[b,d] = sum_e hidden[b,e] * w_proj[e,d]     (64x256 @ 256x256)
// One wave per 16x16 output tile, V_WMMA_F32_16X16X4_F32, K-loop of 64.
// A (16x4 f32): lanes 0-15 -> M=lane, K = k+0,k+1 ; lanes 16-31 -> K = k+2,k+3
// B (4x16 f32): lanes 0-15 -> N=lane, K = k+0,k+1 ; lanes 16-31 -> K = k+2,k+3
// D (16x16 f32): VGPR r: lanes 0-15 -> (M=r, N=lane); lanes 16-31 -> (M=8+r)
// ---------------------------------------------------------------------------
__global__ void __launch_bounds__(32)
hw_gemm_kernel(const float* __restrict__ hidden,
               const float* __restrict__ w_proj,
               float* __restrict__ hw) {
  const int lane = threadIdx.x;
  const int half = lane >> 4;
  const int l15  = lane & 15;
  const int tileN = blockIdx.x;              // d tile: 0..15
  const int tileM = blockIdx.y;              // b tile: 0..3
  const int row = tileM * 16 + l15;          // b row for A
  const int col = tileN * 16 + l15;          // d col for B
  const int koff = half * 2;

  v8f c = {};
#pragma unroll 8
  for (int k = 0; k < D_; k += 4) {
    v2f a, b;
    a.x = hidden[row * D_ + k + koff];
    a.y = hidden[row * D_ + k + koff + 1];
    b.x = w_proj[(k + koff) * D_ + col];
    b.y = w_proj[(k + koff + 1) * D_ + col];
    c = __builtin_amdgcn_wmma_f32_16x16x4_f32(false, a, false, b,
                                              (short)0, c, false, false);
  }
  const int outRow = tileM * 16 + half * 8;
#pragma unroll
  for (int r = 0; r < 8; ++r)
    hw[(outRow + r) * D_ + tileN * 16 + l15] = c[r];
}

// ---------------------------------------------------------------------------
// Kernel 2: bilinear[b,m] = sum_d emb[b,m,d] * hw[b,d]; add recency, mask.
// One wave per (b, 16-row m tile). hw row staged in LDS. B operand has only
// column N=0 populated -> D column 0 holds the 16 dot products.
// ---------------------------------------------------------------------------
__global__ void __launch_bounds__(32)
bilinear_kernel(const float* __restrict__ emb,
                const float* __restrict__ hw,
                const int* __restrict__ last_seen,
                const int* __restrict__ num_emb,
                const float* __restrict__ dscalar,
                const int* __restrict__ timestep,
                float* __restrict__ out_logits,
                float* __restrict__ out_mask) {
  __shared__ float hwS[D_];
  const int lane = threadIdx.x;
  const int half = lane >> 4;
  const int l15  = lane & 15;
  const int m0 = blockIdx.x * 16;
  const int b  = blockIdx.y;

#pragma unroll
  for (int i = 0; i < D_ / 32; ++i)
    hwS[lane + 32 * i] = hw[b * D_ + lane + 32 * i];
  __syncthreads();

  const float* arow = emb + ((size_t)b * M_ + (size_t)(m0 + l15)) * D_;
  const int koff = half * 2;

  v8f c = {};
#pragma unroll 4
  for (int k = 0; k < D_; k += 4) {
    v2f a, bb;
    a.x = arow[k + koff];
    a.y = arow[k + koff + 1];
    float b0 = hwS[k + koff];
    float b1 = hwS[k + koff + 1];
    bb.x = (l15 == 0) ? b0 : 0.0f;     // only column N=0 of B is live
    bb.y = (l15 == 0) ? b1 : 0.0f;
    c = __builtin_amdgcn_wmma_f32_16x16x4_f32(false, a, false, bb,
                                              (short)0, c, false, false);
  }

  // Column N=0: lane 0 holds rows m0..m0+7 in c[0..7]; lane 16 holds m0+8..15
  if (l15 == 0) {
    const float ds = dscalar[0];
    const float ts = (float)timestep[0];
    const int ne = num_emb[b];
#pragma unroll
    for (int r = 0; r < 8; ++r) {
      int m = m0 + half * 8 + r;
      float dist = expf(ds * ((float)last_seen[b * M_ + m] - ts));
      float logit = c[r] + dist;
      bool valid = m < ne;
      out_logits[(size_t)b * M_ + m] = valid ? logit : NEG_MASK_VAL;
      out_mask[(size_t)b * M_ + m]   = valid ? 1.0f : 0.0f;
    }
  }
}

// ---------------------------------------------------------------------------
// Kernel 3: allennlp-style masked log-softmax + target gather, one block / b.
// log_probs = log_softmax(logits + log(mask + 1e-45)); loss = log_probs[tgt]
// ---------------------------------------------------------------------------
__global__ void __launch_bounds__(256)
loss_kernel(const float* __restrict__ logits,
            const float* __restrict__ maskf,
            const int* __restrict__ target,
            float* __restrict__ loss) {
  __shared__ float red[256];
  const int b = blockIdx.x;
  const int tid = threadIdx.x;
  const float logeps = logf(LOG_MASK_EPS);   // log(1e-45); log(1+eps)==0

  float x[M_ / 256];
  float mx = -INFINITY;
#pragma unroll
  for (int i = 0; i < M_ / 256; ++i) {
    int e = tid + 256 * i;
    float xm = logits[(size_t)b * M_ + e] +
               (maskf[(size_t)b * M_ + e] > 0.5f ? 0.0f : logeps);
    x[i] = xm;
    mx = fmaxf(mx, xm);
  }
  red[tid] = mx;
  __syncthreads();
  for (int s = 128; s > 0; s >>= 1) {
    if (tid < s) red[tid] = fmaxf(red[tid], red[tid + s]);
    __syncthreads();
  }
  float gmax = red[0];
  __syncthreads();

  float sum = 0.0f;
#pragma unroll
  for (int i = 0; i < M_ / 256; ++i) sum += expf(x[i] - gmax);
  red[tid] = sum;
  __syncthreads();
  for (int s = 128; s > 0; s >>= 1) {
    if (tid < s) red[tid] += red[tid + s];
    __syncthreads();
  }
  if (tid == 0) {
    int t = target[b];
    float xt = logits[(size_t)b * M_ + t] +
               (maskf[(size_t)b * M_ + t] > 0.5f ? 0.0f : logeps);
    loss[b] = xt - (gmax + logf(red[0]));
  }
}

// ---------------------------------------------------------------------------
extern "C" void kernel_launch(void* const* d_in, const int* in_sizes, int n_in,
                              void* d_out, int out_size, void* d_ws, size_t ws_size,
                              hipStream_t stream) {
  const float* hidden    = (const float*)d_in[0];   // (B, D)
  const float* emb       = (const float*)d_in[1];   // (B, M, D)
  const int*   last_seen = (const int*)d_in[2];     // (B, M)
  const int*   num_emb   = (const int*)d_in[3];     // (B,)
  const int*   target    = (const int*)d_in[4];     // (B,)
  const float* w_proj    = (const float*)d_in[5];   // (D, D)
  const float* dscalar   = (const float*)d_in[6];   // (1,)
  const int*   timestep  = (const int*)d_in[7];     // scalar

  float* out_logits = (float*)d_out;                       // B*M
  float* out_mask   = out_logits + (size_t)B_ * M_;        // B*M
  float* out_loss   = out_mask + (size_t)B_ * M_;          // B
  float* hw         = (float*)d_ws;                        // B*D fp32 scratch

  hipLaunchKernelGGL(hw_gemm_kernel, dim3(D_ / 16, B_ / 16), dim3(32), 0, stream,
                     hidden, w_proj, hw);
  hipLaunchKernelGGL(bilinear_kernel, dim3(M_ / 16, B_), dim3(32), 0, stream,
                     emb, hw, last_seen, num_emb, dscalar, timestep,
                     out_logits, out_mask);
  hipLaunchKernelGGL(loss_kernel, dim3(B_), dim3(256), 0, stream,
                     out_logits, out_mask, target, out_loss);
}